// Eland_Model_41008347742351
// MI455X (gfx1250) — compile-verified
//
#include <hip/hip_runtime.h>
#include <hip/hip_bf16.h>
#include <float.h>

typedef __attribute__((ext_vector_type(2))) float v2f;
typedef __attribute__((ext_vector_type(4))) float v4f;
typedef __attribute__((ext_vector_type(8))) float v8f;

#define BATCH   1024
#define FEAT    256
#define DICT    100000
#define NUSERS  2048
#define NTILES  (DICT / 16)                       // 6250 exact
#define NCHUNK  64
#define TPC     ((NTILES + NCHUNK - 1) / NCHUNK)  // 98
#define MROWS   32                                // M rows per GEMM block (2 subtiles)
#define MBLKS   (BATCH / MROWS)                   // 32
#define NGEMM   (NCHUNK * MBLKS)                  // 2048 GEMM blocks
#define NCOPY   4096                              // copy blocks (interleaved 2-of-3)
#define LDSW    260                               // padded row stride (words) -> conflict-free
#define EPSF    1e-8f

// ---------------------------------------------------------------------------
// Fused kernel: bid%3==0 -> GEMM+argmax block; bid%3!=0 -> adjacency copy
// block. Interleaving keeps HBM-streaming copy blocks co-resident with the
// L2-bound WMMA blocks so the 1.64 GB adjacency stream hides under the GEMM.
// GEMM block: 32 delta rows in LDS, two 16x16 f32 accumulators per wave share
// each B fragment (halves idx2feats L2 traffic vs M=16 blocking).
// ---------------------------------------------------------------------------
__global__ __launch_bounds__(256) void fused_gemm_copy_kernel(
    const float* __restrict__ delta,
    const float* __restrict__ idx2feats,
    const float* __restrict__ gumbel,
    const float* __restrict__ adj,
    float* __restrict__ out,
    float* __restrict__ pVal, int* __restrict__ pIdx)
{
    const int bid  = blockIdx.x;
    const int role = bid % 3;
    const int tid  = threadIdx.x;

    if (role != 0) {
        // ---------------- streaming adjacency copy (non-temporal B128) -----
        const size_t ADJ4   = (size_t)NUSERS * DICT / 4;     // 51,200,000
        const size_t copyId = (size_t)(bid / 3) * 2 + (role - 1);   // 0..NCOPY-1
        const size_t stride = (size_t)NCOPY * 256;
        for (size_t id = copyId * 256 + tid; id < ADJ4; id += stride) {
            v4f v = __builtin_nontemporal_load(((const v4f*)adj) + id);
            __builtin_nontemporal_store(v, ((v4f*)out) + id);
        }
        return;
    }

    // -------------------- GEMM + fused gumbel argmax -----------------------
    __shared__ float ldsA[MROWS * LDSW];
    __shared__ float ldsN1[MROWS];
    __shared__ float wVal[8][MROWS];
    __shared__ int   wIdx[8][MROWS];

    const int g     = bid / 3;            // 0..NGEMM-1
    const int chunk = g & (NCHUNK - 1);
    const int mBase = (g >> 6) * MROWS;
    const int lane  = tid & 31;
    const int wave  = tid >> 5;
    const int half  = lane >> 4;          // 0: lanes 0-15, 1: lanes 16-31
    const int l16   = lane & 15;

    // cooperative load of A-tile (32 rows x 256 K) into padded LDS
    {
        const v4f* src = (const v4f*)(delta + (size_t)mBase * FEAT);
        #pragma unroll
        for (int i = 0; i < 8; ++i) {
            int f   = tid + i * 256;      // float4 index 0..2047
            int row = f >> 6;
            int c4  = f & 63;
            v4f v = src[f];
            *(v4f*)&ldsA[row * LDSW + c4 * 4] = v;  // LDSW*4 = 1040 B = 65*16 -> 16B aligned
        }
    }
    __syncthreads();

    // delta row norms: waves 0,1 each cover 16 rows (lane halves split K)
    if (wave < 2) {
        float s = 0.f;
        const float* rp = &ldsA[(wave * 16 + l16) * LDSW + half * 128];
        #pragma unroll
        for (int k = 0; k < 128; k += 4) {
            v4f v = *(const v4f*)(rp + k);
            s += v.x * v.x + v.y * v.y + v.z * v.z + v.w * v.w;
        }
        s += __shfl_xor(s, 16, 32);
        if (half == 0) ldsN1[wave * 16 + l16] = sqrtf(s);
    }
    __syncthreads();

    float n1v[2][8];
    #pragma unroll
    for (int s = 0; s < 2; ++s)
        #pragma unroll
        for (int r = 0; r < 8; ++r)
            n1v[s][r] = ldsN1[s * 16 + r + 8 * half];   // m = s*16 + r + 8*half

    float runV[2][8]; int runI[2][8];
    #pragma unroll
    for (int s = 0; s < 2; ++s)
        #pragma unroll
        for (int r = 0; r < 8; ++r) { runV[s][r] = -FLT_MAX; runI[s][r] = 0; }

    const int tStart = chunk * TPC;
    const int tEnd   = min(tStart + TPC, NTILES);

    for (int t = tStart + wave; t < tEnd; t += 8) {
        const int n0 = t * 16;
        // B-matrix 4x16 f32 layout: lane l -> n = l%16, K pair base = 2*(l/16)
        const float* brow  = idx2feats + (size_t)(n0 + l16) * FEAT + 2 * half;
        // A-matrix 16x4 f32 layout: lane l -> m = l%16 (per subtile), K pair base = 2*(l/16)
        const float* a0row = &ldsA[l16 * LDSW + 2 * half];
        const float* a1row = &ldsA[(16 + l16) * LDSW + 2 * half];

        v8f acc[2] = {};
        float bsq = 0.f;
        #pragma unroll 4
        for (int kb = 0; kb < 64; ++kb) {
            v2f b  = *(const v2f*)(brow  + kb * 4);
            v2f a0 = *(const v2f*)(a0row + kb * 4);
            v2f a1 = *(const v2f*)(a1row + kb * 4);
            bsq += b.x * b.x + b.y * b.y;
            acc[0] = __builtin_amdgcn_wmma_f32_16x16x4_f32(
                         false, a0, false, b, (short)0, acc[0], false, false);
            acc[1] = __builtin_amdgcn_wmma_f32_16x16x4_f32(
                         false, a1, false, b, (short)0, acc[1], false, false);
        }
        // dict-row norm: lane and lane^16 together cover all 256 K of column n
        float n2 = sqrtf(bsq + __shfl_xor(bsq, 16, 32));

        const int n = n0 + l16;
        #pragma unroll
        for (int s = 0; s < 2; ++s) {
            const float* g0 = gumbel + (size_t)(mBase + s * 16 + 8 * half) * DICT + n;
            #pragma unroll
            for (int r = 0; r < 8; ++r) {
                float denom = n1v[s][r] * n2 + EPSF;
                float val   = acc[s][r] / denom + g0[(size_t)r * DICT];
                if (val > runV[s][r]) { runV[s][r] = val; runI[s][r] = n; }  // ascending n -> first-hit ties
            }
        }
    }

    // argmax over the 16 lanes of each half (butterfly)
    #pragma unroll
    for (int off = 1; off < 16; off <<= 1) {
        #pragma unroll
        for (int s = 0; s < 2; ++s)
            #pragma unroll
            for (int r = 0; r < 8; ++r) {
                float ov = __shfl_xor(runV[s][r], off, 32);
                int   oi = __shfl_xor(runI[s][r], off, 32);
                if (ov > runV[s][r] || (ov == runV[s][r] && oi < runI[s][r])) {
                    runV[s][r] = ov; runI[s][r] = oi;
                }
            }
    }
    if (l16 == 0) {
        #pragma unroll
        for (int s = 0; s < 2; ++s)
            #pragma unroll
            for (int r = 0; r < 8; ++r) {
                wVal[wave][s * 16 + r + 8 * half] = runV[s][r];
                wIdx[wave][s * 16 + r + 8 * half] = runI[s][r];
            }
    }
    __syncthreads();

    if (tid < MROWS) {
        float bv = wVal[0][tid]; int bi = wIdx[0][tid];
        #pragma unroll
        for (int w = 1; w < 8; ++w) {
            float v = wVal[w][tid]; int i = wIdx[w][tid];
            if (v > bv || (v == bv && i < bi)) { bv = v; bi = i; }
        }
        pVal[(size_t)chunk * BATCH + mBase + tid] = bv;
        pIdx[(size_t)chunk * BATCH + mBase + tid] = bi;
    }
}

// ---------------------------------------------------------------------------
// Kernel 2: reduce chunk partials -> top_idx[1024] (ties -> lower index)
// ---------------------------------------------------------------------------
__global__ void reduce_kernel(const float* __restrict__ pVal,
                              const int* __restrict__ pIdx,
                              int* __restrict__ topIdx)
{
    int m = blockIdx.x * blockDim.x + threadIdx.x;
    if (m >= BATCH) return;
    float bv = -FLT_MAX; int bi = 0;
    for (int c = 0; c < NCHUNK; ++c) {
        float v = pVal[(size_t)c * BATCH + m];
        int   i = pIdx[(size_t)c * BATCH + m];
        if (v > bv || (v == bv && i < bi)) { bv = v; bi = i; }
    }
    topIdx[m] = bi;
}

// ---------------------------------------------------------------------------
// Kernel 3: pred_features gather (1 MB, after argmax is known)
// ---------------------------------------------------------------------------
__global__ void gather_kernel(const float* __restrict__ idx2feats,
                              const int* __restrict__ topIdx,
                              float* __restrict__ out)
{
    const size_t ADJ4 = (size_t)NUSERS * DICT / 4;
    int id = blockIdx.x * blockDim.x + threadIdx.x;   // float4 index
    if (id >= BATCH * FEAT / 4) return;
    int row = id >> 6;
    int c4  = id & 63;
    v4f v = ((const v4f*)(idx2feats + (size_t)topIdx[row] * FEAT))[c4];
    ((v4f*)out)[ADJ4 + id] = v;
}

// ---------------------------------------------------------------------------
// Kernel 4: duplicate-safe scatter-add of mask * one_hot(top_idx).
// Adding 1.0f repeatedly is exact -> order-independent -> deterministic.
// ---------------------------------------------------------------------------
__global__ void scatter_kernel(const int* __restrict__ indices,
                               const unsigned char* __restrict__ mask,
                               const int* __restrict__ topIdx,
                               float* __restrict__ outAdj)
{
    int i = blockIdx.x * blockDim.x + threadIdx.x;
    if (i >= BATCH) return;
    if (mask[i])
        atomicAdd(outAdj + (size_t)indices[i] * DICT + topIdx[i], 1.0f);
}

extern "C" void kernel_launch(void* const* d_in, const int* in_sizes, int n_in,
                              void* d_out, int out_size, void* d_ws, size_t ws_size,
                              hipStream_t stream)
{
    const float* delta         = (const float*)d_in[0];
    const float* idx2feats     = (const float*)d_in[1];
    const float* original_adj  = (const float*)d_in[2];
    const float* gumbel        = (const float*)d_in[3];
    const int*   indices       = (const int*)d_in[4];
    const unsigned char* mask  = (const unsigned char*)d_in[5];   // jnp.bool_ = 1 byte
    float* out = (float*)d_out;

    // workspace layout: partial values, partial indices, top_idx (~528 KB)
    float* pVal   = (float*)d_ws;
    int*   pIdx   = (int*)(pVal + (size_t)NCHUNK * BATCH);
    int*   topIdx = pIdx + (size_t)NCHUNK * BATCH;

    const int totalBlocks = NGEMM + NCOPY;   // 2048 GEMM + 4096 copy, interleaved 1:2
    fused_gemm_copy_kernel<<<totalBlocks, 256, 0, stream>>>(
        delta, idx2feats, gumbel, original_adj, out, pVal, pIdx);
    reduce_kernel<<<(BATCH + 255) / 256, 256, 0, stream>>>(pVal, pIdx, topIdx);
    gather_kernel<<<(BATCH * FEAT / 4 + 255) / 256, 256, 0, stream>>>(idx2feats, topIdx, out);
    scatter_kernel<<<(BATCH + 255) / 256, 256, 0, stream>>>(indices, mask, topIdx, out);
}